// NeuralDictionaryV6BDouble_38594576121950
// MI455X (gfx1250) — compile-verified
//
#include <hip/hip_runtime.h>
#include <hip/hip_bf16.h>

// ---------------------------------------------------------------------------
// NeuralDictionary forward for MI455X (gfx1250, wave32).
//   scores = -L1(keys, q); softmax; out = sigmoid(softmax @ values)
// HBM-bound (~1 GB traffic, ~45us floor at 23.3 TB/s):
//   - keys (512MB) streamed once in k_scores with NT b128 loads
//   - values (512MB) streamed once in k_wsum with NT loads
//   - 1MB score vector round-trips through L2 (RT path)
// Softmax weights applied to V via V_WMMA_F32_16X16X4_F32 rank-4 updates.
// All reductions are fixed-order (no float atomics) -> bit-deterministic.
// ---------------------------------------------------------------------------

typedef __attribute__((ext_vector_type(2))) float v2f;
typedef __attribute__((ext_vector_type(4))) float v4f;   // native clang vector
typedef __attribute__((ext_vector_type(8))) float v8f;

#define CAP   262144
#define NF    512
#define K1_BLOCKS (CAP / 8)      // 32768 blocks, 8 rows (waves) each
#define K3_BLOCKS (CAP / 1024)   // 256 blocks, 1024 rows each

// Workspace layout (float offsets). Total ~426k floats (~1.7 MB).
#define SCORES_OFF 0
#define PMAX_OFF   (SCORES_OFF + CAP)        // 262144 : per-block score max
#define M_OFF      (PMAX_OFF + K1_BLOCKS)    // 294912 : global max
#define PSUM_OFF   (M_OFF + 16)              // 294928 : per-block sum(exp)
#define PACC_OFF   (PSUM_OFF + K3_BLOCKS)    // 295184 : per-block out partials

// ---------------- Kernel 1: scores + per-block max (streams keys, NT) ------
__global__ __launch_bounds__(256) void k_scores(const float* __restrict__ q,
                                                const float* __restrict__ keys,
                                                float* __restrict__ ws) {
  __shared__ float sq[NF];
  __shared__ float wmax[8];
  const int tid = threadIdx.x;
  for (int j = tid; j < NF; j += 256) sq[j] = q[j];
  __syncthreads();

  const int wave = tid >> 5, lane = tid & 31;
  const int row  = blockIdx.x * 8 + wave;          // one row per wave
  const v4f* krow = (const v4f*)(keys + (size_t)row * NF);
  const v4f* q4   = (const v4f*)sq;

  float s = 0.f;
#pragma unroll
  for (int t = 0; t < 4; ++t) {                    // 4 x global_load_b128 NT
    const int idx = t * 32 + lane;
    v4f k4 = __builtin_nontemporal_load(&krow[idx]);
    v4f qq = q4[idx];
    s += fabsf(k4.x - qq.x) + fabsf(k4.y - qq.y) +
         fabsf(k4.z - qq.z) + fabsf(k4.w - qq.w);
  }
#pragma unroll
  for (int off = 16; off >= 1; off >>= 1)          // wave32 butterfly reduce
    s += __shfl_xor(s, off, 32);

  const float score = -s;
  if (lane == 0) { ws[SCORES_OFF + row] = score; wmax[wave] = score; }
  __syncthreads();
  if (tid == 0) {
    float m = wmax[0];
#pragma unroll
    for (int i = 1; i < 8; ++i) m = fmaxf(m, wmax[i]);
    ws[PMAX_OFF + blockIdx.x] = m;
  }
}

// ---------------- Kernel 2: global max over per-block maxes ----------------
__global__ __launch_bounds__(1024) void k_max(float* __restrict__ ws) {
  __shared__ float red[1024];
  const int tid = threadIdx.x;
  float m = -__builtin_inff();
  for (int i = tid; i < K1_BLOCKS; i += 1024) m = fmaxf(m, ws[PMAX_OFF + i]);
  red[tid] = m;
  __syncthreads();
  for (int s = 512; s > 0; s >>= 1) {
    if (tid < s) red[tid] = fmaxf(red[tid], red[tid + s]);
    __syncthreads();
  }
  if (tid == 0) ws[M_OFF] = red[0];
}

// ---------------- Kernel 3: exp weights + WMMA value accumulation ----------
// Block = 1024 threads (32 waves) owns 1024 rows of V.
// Wave w owns output columns [16w, 16w+16). Each V_WMMA_F32_16X16X4_F32
// performs out[n] += sum_{k<4} w_k * V[k][n]:
//   A (16x4): every row = (w0..w3)  -> lanes 0-15 hold {w0,w1}, 16-31 {w2,w3}
//   B (4x16): lanes 0-15 hold {V[r][n],V[r+1][n]}, lanes 16-31 rows r+2,r+3
//   D rows are all identical; lane<16 VGPR0 = column n result.
__global__ __launch_bounds__(1024) void k_wsum(const float* __restrict__ values,
                                               float* __restrict__ ws) {
  __shared__ float w[1024];
  __shared__ float red[1024];
  const int tid  = threadIdx.x;
  const int base = blockIdx.x * 1024;

  // Phase A: softmax numerators for this block's 1024 rows + partial sum.
  const float m  = ws[M_OFF];
  const float wi = __expf(ws[SCORES_OFF + base + tid] - m);
  w[tid]   = wi;
  red[tid] = wi;
  __syncthreads();
  for (int s = 512; s > 0; s >>= 1) {
    if (tid < s) red[tid] += red[tid + s];
    __syncthreads();
  }
  if (tid == 0) ws[PSUM_OFF + blockIdx.x] = red[0];

  // Phase B: rank-4 WMMA updates over 1024 rows (EXEC all-ones, no divergence).
  const int wave = tid >> 5, lane = tid & 31;
  const int col  = wave * 16 + (lane & 15);
  const int koff = (lane >= 16) ? 2 : 0;
  const float* __restrict__ vc = values + col;

  v8f c = {};
  for (int k = 0; k < 1024; k += 4) {
    const int r = k + koff;
    v2f a, b;
    a.x = w[r];
    a.y = w[r + 1];
    b.x = __builtin_nontemporal_load(&vc[(size_t)(base + r)     * NF]);
    b.y = __builtin_nontemporal_load(&vc[(size_t)(base + r + 1) * NF]);
    c = __builtin_amdgcn_wmma_f32_16x16x4_f32(
        /*neg_a=*/false, a, /*neg_b=*/false, b,
        /*c_mod=*/(short)0, c, /*reuse_a=*/false, /*reuse_b=*/false);
  }
  // D row M=0 lives in VGPR0 of lanes 0-15 (all rows equal by construction).
  if (lane < 16) ws[PACC_OFF + blockIdx.x * NF + col] = c[0];
}

// ---------------- Kernel 4: fixed-order final reduce + sigmoid -------------
__global__ __launch_bounds__(512) void k_final(const float* __restrict__ ws,
                                               float* __restrict__ out) {
  const int n = threadIdx.x;
  float sum = 0.f;
  for (int b = 0; b < K3_BLOCKS; ++b) sum += ws[PSUM_OFF + b];
  float acc = 0.f;
  for (int b = 0; b < K3_BLOCKS; ++b) acc += ws[PACC_OFF + b * NF + n];
  const float x = acc / sum;
  out[n] = 1.f / (1.f + __expf(-x));
}

// ---------------------------------------------------------------------------
extern "C" void kernel_launch(void* const* d_in, const int* in_sizes, int n_in,
                              void* d_out, int out_size, void* d_ws, size_t ws_size,
                              hipStream_t stream) {
  const float* q      = (const float*)d_in[0];   // [512]
  const float* keys   = (const float*)d_in[1];   // [262144, 512]
  const float* values = (const float*)d_in[2];   // [262144, 512]
  float* out = (float*)d_out;                    // [512]
  float* ws  = (float*)d_ws;

  k_scores<<<K1_BLOCKS, 256, 0, stream>>>(q, keys, ws);
  k_max   <<<1,        1024, 0, stream>>>(ws);
  k_wsum  <<<K3_BLOCKS, 1024, 0, stream>>>(values, ws);
  k_final <<<1,         512, 0, stream>>>(ws, out);
}